// LowRankSS2D_7911329759952
// MI455X (gfx1250) — compile-verified
//
#include <hip/hip_runtime.h>
#include <hip/hip_bf16.h>
#include <math.h>

// ---------------------------------------------------------------------------
// LowRankSS2D on MI455X (gfx1250, wave32).
//
// Rank-64 collapse of the SSM recurrence:
//   h_t = (U V^T) h_{t-1} + b_t  ==>  g_{t+1} = W g_t + c_t,  h_t = b_t + U g_t
//   with W = V^T U (64x64), c_t = V^T b_t.
// All position-wise work is 16x16-tile GEMM via V_WMMA_F32_16X16X4_F32.
// The recurrence itself is *blocked* (Q=16): chunk carries via GEMM (phase A),
// a 256-step boundary scan (phase B), and full interior reconstruction as one
// big triangular GEMM (phase C) — sequential depth 4096 -> 256, remainder is
// device-wide WMMA work against an L2-resident 4.5MB combine matrix.
// ---------------------------------------------------------------------------

typedef float v2f __attribute__((ext_vector_type(2)));
typedef float v8f __attribute__((ext_vector_type(8)));

// A (16x4 f32): lane = m + 16*g holds A[m,k0+2g], A[m,k0+2g+1].
// B (4x16):     lane = n + 16*g holds B[k0+2g,n], B[k0+2g+1,n].
// C/D (16x16):  vgpr v, lane = n + 16*gg holds D[v+8*gg, n].
__device__ __forceinline__ v8f wmma4(v2f a, v2f b, v8f c) {
  return __builtin_amdgcn_wmma_f32_16x16x4_f32(false, a, false, b,
                                               (short)0, c, false, false);
}

__device__ __forceinline__ v8f gemm_tile(const float* __restrict__ sA, int lds,
                                         const float* __restrict__ B, int ldb,
                                         int n0, int K, int lane) {
  const int nn = lane & 15;
  const int g  = lane >> 4;
  v8f acc = {};
  for (int k0 = 0; k0 < K; k0 += 4) {
    v2f a, b;
    a.x = sA[nn * lds + k0 + 2 * g];
    a.y = sA[nn * lds + k0 + 2 * g + 1];
    b.x = B[(k0 + 2 * g) * ldb + n0 + nn];
    b.y = B[(k0 + 2 * g + 1) * ldb + n0 + nn];
    acc = wmma4(a, b, acc);
  }
  return acc;
}

// ---------------------------------------------------------------------------
// Prep kernels
// ---------------------------------------------------------------------------
__global__ void k_transpose(float* __restrict__ dst, const float* __restrict__ src,
                            int R, int Cc, int n) {
  int i = blockIdx.x * blockDim.x + threadIdx.x;
  if (i < n) {
    int r = i / Cc, c = i - r * Cc;
    dst[c * R + r] = src[i];
  }
}

// Wt[k][n] = W[n][k] = sum_s V[s,n]*U[s,k]
__global__ void k_wt(const float* __restrict__ U, const float* __restrict__ V,
                     float* __restrict__ Wt) {
  int i = blockIdx.x * 256 + threadIdx.x;
  int k = i >> 6, n = i & 63;
  float s = 0.f;
  for (int ss = 0; ss < 128; ++ss) s += V[ss * 64 + n] * U[ss * 64 + k];
  Wt[i] = s;
}

// Pw[j] = W^j (row-major), j = 0..16. One block; 16 tiny chained GEMMs.
__global__ void __launch_bounds__(256) k_powers(const float* __restrict__ Wt,
                                                float* __restrict__ Pw) {
  __shared__ float sP[64 * 64];
  int tid = threadIdx.x;
  for (int i = tid; i < 4096; i += 256) {
    float v = ((i >> 6) == (i & 63)) ? 1.f : 0.f;
    Pw[i] = v;
    sP[i] = v;
  }
  __syncthreads();
  for (int j = 1; j <= 16; ++j) {
    float tmp[16];
#pragma unroll
    for (int e = 0; e < 16; ++e) {
      int i = tid * 16 + e;
      int r = i >> 6, rp = i & 63;
      float s = 0.f;
      for (int k = 0; k < 64; ++k) s += sP[r * 64 + k] * Wt[rp * 64 + k];
      tmp[e] = s;
    }
    __syncthreads();
#pragma unroll
    for (int e = 0; e < 16; ++e) {
      int i = tid * 16 + e;
      sP[i] = tmp[e];
      Pw[j * 4096 + i] = tmp[e];
    }
    __syncthreads();
  }
}

// Te[(i*64+r')*64 + r] = (W^{15-i})[r][r']  -> E = C_chunk @ Te
__global__ void k_build_te(const float* __restrict__ Pw, float* __restrict__ Te) {
  int i = blockIdx.x * 256 + threadIdx.x;   // 65536
  int krow = i >> 6, r = i & 63;
  int ii = krow >> 6, rp = krow & 63;
  Te[i] = Pw[(15 - ii) * 4096 + r * 64 + rp];
}

// M2 (1088 x 1024): rows 0..63 = W^j blocks (G propagation), rows 64.. =
// triangular prefix blocks (W^{j-1-i})^T, zero for i >= j.
__global__ void k_build_m2(const float* __restrict__ Pw, float* __restrict__ M2) {
  int i = blockIdx.x * 256 + threadIdx.x;   // 1088*1024 = 1114112
  int k = i >> 10, n = i & 1023;
  int j = n >> 6, r = n & 63;
  float v;
  if (k < 64) {
    v = Pw[j * 4096 + r * 64 + k];
  } else {
    int k2 = k - 64;
    int ii = k2 >> 6, rp = k2 & 63;
    v = (ii < j) ? Pw[(j - 1 - ii) * 4096 + r * 64 + rp] : 0.f;
  }
  M2[i] = v;
}

// ---------------------------------------------------------------------------
// k_chain: seq -> xs (-> state_seq) -> bx (-> ws) -> c = bx@V (-> ws)
// ---------------------------------------------------------------------------
__global__ void __launch_bounds__(128) k_chain(
    const float* __restrict__ x, const float* __restrict__ in_wT,
    const float* __restrict__ in_b, const float* __restrict__ B_wT,
    const float* __restrict__ B_b, const float* __restrict__ V,
    float* __restrict__ state_out, float* __restrict__ bx_ws,
    float* __restrict__ c_ws) {
  const int L = 4096, C = 192, S = 128, R = 64, H = 64, W = 64;
  const int lt = blockIdx.x, b = blockIdx.y, o = blockIdx.z;
  const int l0 = lt * 16;
  const int tid = threadIdx.x, lane = tid & 31, wv = tid >> 5;
  const int nn = lane & 15, g = lane >> 4;
  __shared__ float sA[16 * 193];
  __shared__ float sX[16 * 129];

  for (int i = tid; i < 16 * C; i += 128) {
    int m = i & 15, c = i >> 4;
    int l = l0 + m;
    int h = o ? (l & 63) : (l >> 6);
    int w = o ? (l >> 6) : (l & 63);
    sA[m * 193 + c] = x[(((size_t)b * C + c) * H + h) * W + w];
  }
  __syncthreads();

  float* bxo = bx_ws + (size_t)o * 8 * L * S;
  float* co  = c_ws  + (size_t)o * 8 * L * R;

  for (int t = 0; t < 2; ++t) {
    int n0 = (wv * 2 + t) * 16;
    v8f acc = gemm_tile(sA, 193, in_wT, S, n0, C, lane);
    float bias = in_b[n0 + nn];
#pragma unroll
    for (int v = 0; v < 8; ++v) {
      int row = v + 8 * g;
      float val = acc[v] + bias;
      sX[row * 129 + n0 + nn] = val;
      if (o == 0)
        state_out[((size_t)(b * L + l0 + row)) * S + n0 + nn] = val;
    }
  }
  __syncthreads();

  for (int t = 0; t < 2; ++t) {
    int n0 = (wv * 2 + t) * 16;
    v8f acc = gemm_tile(sX, 129, B_wT, S, n0, S, lane);
    float bias = B_b[n0 + nn];
#pragma unroll
    for (int v = 0; v < 8; ++v) {
      int row = v + 8 * g;
      float val = acc[v] + bias;
      sA[row * 193 + n0 + nn] = val;
      bxo[((size_t)(b * L + l0 + row)) * S + n0 + nn] = val;
    }
  }
  __syncthreads();

  {
    int n0 = wv * 16;
    v8f acc = gemm_tile(sA, 193, V, R, n0, S, lane);
#pragma unroll
    for (int v = 0; v < 8; ++v) {
      int row = v + 8 * g;
      co[((size_t)(b * L + l0 + row)) * R + n0 + nn] = acc[v];
    }
  }
}

// ---------------------------------------------------------------------------
// Phase A: chunk carries E[d,b,q] = C_chunk @ Te.  512 blocks x 128 thr.
// Row tile = 16 chunks (q0..q0+15) of one (d,b).
// ---------------------------------------------------------------------------
__global__ void __launch_bounds__(128) k_carry_e(const float* __restrict__ cbuf,
                                                 const float* __restrict__ Te,
                                                 float* __restrict__ E) {
  const int L = 4096, Bn = 8, R = 64, NQ = 256, Q = 16;
  int rr0 = blockIdx.x * 16;
  int d = rr0 >> 11, rem = rr0 & 2047, b = rem >> 8, q0 = rem & 255;
  int pair = d >> 1, bwd = d & 1;
  const float* cb = cbuf + (size_t)pair * Bn * L * R;
  float* Eo = E + (size_t)d * Bn * NQ * R;
  int tid = threadIdx.x, lane = tid & 31, wv = tid >> 5;
  int nn = lane & 15, g = lane >> 4;
  int n0 = wv * 16;
  __shared__ float sA[16 * 65];
  v8f acc = {};
  for (int kc = 0; kc < Q; ++kc) {
    __syncthreads();
    for (int i = tid; i < 16 * 64; i += 128) {
      int m = i >> 6, rp = i & 63;
      int tloc = (q0 + m) * Q + kc;
      int l = bwd ? (L - 1 - tloc) : tloc;
      sA[m * 65 + rp] = cb[((size_t)(b * L + l)) * 64 + rp];
    }
    __syncthreads();
#pragma unroll
    for (int kk = 0; kk < 16; ++kk) {
      v2f a, bf;
      a.x = sA[nn * 65 + 4 * kk + 2 * g];
      a.y = sA[nn * 65 + 4 * kk + 2 * g + 1];
      int krow = kc * 64 + 4 * kk + 2 * g;
      bf.x = Te[(size_t)krow * 64 + n0 + nn];
      bf.y = Te[(size_t)(krow + 1) * 64 + n0 + nn];
      acc = wmma4(a, bf, acc);
    }
  }
#pragma unroll
  for (int v = 0; v < 8; ++v) {
    int row = v + 8 * g;
    Eo[((size_t)(b * NQ + q0 + row)) * 64 + n0 + nn] = acc[v];
  }
}

// ---------------------------------------------------------------------------
// Phase B: boundary scan G[q+1] = W^Q G[q] + E[q], 256 steps. 4 blocks.
// W^Q fragments live in VGPRs for the whole loop; G ping-pongs through LDS.
// ---------------------------------------------------------------------------
__global__ void __launch_bounds__(128) k_carry_scan(const float* __restrict__ E,
                                                    const float* __restrict__ PwQ,
                                                    float* __restrict__ Gq) {
  const int Bn = 8, R = 64, NQ = 256;
  int d = blockIdx.x;
  int tid = threadIdx.x, lane = tid & 31, wv = tid >> 5;
  int nn = lane & 15, g = lane >> 4;
  int n0 = wv * 16;
  __shared__ float sG[16 * 65];
  v2f wfrag[16];
#pragma unroll
  for (int kk = 0; kk < 16; ++kk) {   // B[k][n] = (W^Q)[n][k]
    wfrag[kk].x = PwQ[(size_t)(n0 + nn) * 64 + 4 * kk + 2 * g];
    wfrag[kk].y = PwQ[(size_t)(n0 + nn) * 64 + 4 * kk + 2 * g + 1];
  }
  for (int i = tid; i < 16 * 65; i += 128) sG[i] = 0.f;
  __syncthreads();
  const float* Eo = E + (size_t)d * Bn * NQ * R;
  float* Go = Gq + (size_t)d * Bn * NQ * R;
  for (int q = 0; q < NQ; ++q) {
    for (int i = tid; i < 8 * 64; i += 128) {
      int bb = i >> 6, r = i & 63;
      Go[((size_t)(bb * NQ + q)) * 64 + r] = sG[bb * 65 + r];
    }
    v8f acc = {};
#pragma unroll
    for (int kk = 0; kk < 16; ++kk) {
      v2f a;
      a.x = sG[nn * 65 + 4 * kk + 2 * g];
      a.y = sG[nn * 65 + 4 * kk + 2 * g + 1];
      acc = wmma4(a, wfrag[kk], acc);
    }
    if (g == 0) {
#pragma unroll
      for (int v = 0; v < 8; ++v)
        acc[v] += Eo[((size_t)(v * NQ + q)) * 64 + n0 + nn];
    }
    __syncthreads();
#pragma unroll
    for (int v = 0; v < 8; ++v) sG[(v + 8 * g) * 65 + n0 + nn] = acc[v];
    __syncthreads();
  }
}

// ---------------------------------------------------------------------------
// Phase C: interior reconstruction g[qQ+j] = [G[q] | C_chunk] @ M2.
// 512 blocks x 256 thr (8 waves); K chunked by 64 through LDS; triangular
// zero blocks skipped (kc <= j); n-tiles interleaved across waves to balance.
// ---------------------------------------------------------------------------
__global__ void __launch_bounds__(256) k_rebuild(const float* __restrict__ cbuf,
                                                 const float* __restrict__ Gq,
                                                 const float* __restrict__ M2,
                                                 float* __restrict__ g4) {
  const int L = 4096, Bn = 8, R = 64, NQ = 256, Q = 16;
  int rr0 = blockIdx.x * 16;
  int d = rr0 >> 11, rem = rr0 & 2047, b = rem >> 8, q0 = rem & 255;
  int pair = d >> 1, bwd = d & 1;
  const float* cb = cbuf + (size_t)pair * Bn * L * R;
  const float* Go = Gq + (size_t)d * Bn * NQ * R;
  float* go = g4 + (size_t)d * Bn * L * R;
  int tid = threadIdx.x, lane = tid & 31, wv = tid >> 5;
  int nn = lane & 15, g = lane >> 4;
  __shared__ float sA[16 * 65];
  v8f zero = {};
  v8f acc[8];
#pragma unroll
  for (int t = 0; t < 8; ++t) acc[t] = zero;

  for (int kc = 0; kc <= Q; ++kc) {
    __syncthreads();
    for (int i = tid; i < 16 * 64; i += 256) {
      int m = i >> 6, rp = i & 63;
      float v;
      if (kc == 0) {
        v = Go[((size_t)(b * NQ + q0 + m)) * 64 + rp];
      } else {
        int tloc = (q0 + m) * Q + (kc - 1);
        int l = bwd ? (L - 1 - tloc) : tloc;
        v = cb[((size_t)(b * L + l)) * 64 + rp];
      }
      sA[m * 65 + rp] = v;
    }
    __syncthreads();
#pragma unroll
    for (int t = 0; t < 8; ++t) {
      int n0 = (t * 8 + wv) * 16;     // interleave j across waves
      int j = n0 >> 6;
      if (kc > j) continue;           // triangular zero block (wave-uniform)
#pragma unroll
      for (int kk = 0; kk < 16; ++kk) {
        v2f a, bf;
        a.x = sA[nn * 65 + 4 * kk + 2 * g];
        a.y = sA[nn * 65 + 4 * kk + 2 * g + 1];
        int krow = kc * 64 + 4 * kk + 2 * g;
        bf.x = M2[(size_t)krow * 1024 + n0 + nn];
        bf.y = M2[(size_t)(krow + 1) * 1024 + n0 + nn];
        acc[t] = wmma4(a, bf, acc[t]);
      }
    }
  }
#pragma unroll
  for (int t = 0; t < 8; ++t) {
    int n0 = (t * 8 + wv) * 16;
    int j = n0 >> 6;
    int r = (n0 & 63) + nn;
#pragma unroll
    for (int v = 0; v < 8; ++v) {
      int row = v + 8 * g;
      int tloc = (q0 + row) * Q + j;
      int l = bwd ? (L - 1 - tloc) : tloc;
      go[((size_t)(b * L + l)) * 64 + r] = acc[t][v];
    }
  }
}

// ---------------------------------------------------------------------------
// k_out: hsum = 0.5*(bx_lr+bx_tb) + 0.25*(sum_d g_d)@U^T; y = hsum@out_w^T+b;
// z = silu(seq@gate_w^T+b)*y + seq; layernorm; NCHW store.
// ---------------------------------------------------------------------------
__global__ void __launch_bounds__(128) k_out(
    const float* __restrict__ x, const float* __restrict__ bx_ws,
    const float* __restrict__ g4, const float* __restrict__ U_T,
    const float* __restrict__ out_wT, const float* __restrict__ out_b,
    const float* __restrict__ gate_wT, const float* __restrict__ gate_b,
    const float* __restrict__ norm_w, const float* __restrict__ norm_b,
    float* __restrict__ y_out) {
  const int L = 4096, C = 192, S = 128, R = 64, H = 64, W = 64, Bn = 8;
  const int lt = blockIdx.x, b = blockIdx.y;
  const int l0 = lt * 16;
  const int tid = threadIdx.x, lane = tid & 31, wv = tid >> 5;
  const int nn = lane & 15, g = lane >> 4;
  __shared__ float sG[16 * 65];
  __shared__ float sH[16 * 129];
  __shared__ float sSeq[16 * 193];
  __shared__ float sZ[16 * 200];
  __shared__ float red1[128], red2[128];
  __shared__ float stats[32];

  for (int i = tid; i < 16 * 64; i += 128) {
    int m = i >> 6, r = i & 63;
    float s = 0.f;
    for (int dd = 0; dd < 4; ++dd)
      s += g4[(size_t)dd * Bn * L * R + ((size_t)(b * L + l0 + m)) * R + r];
    sG[m * 65 + r] = s;
  }
  for (int i = tid; i < 16 * C; i += 128) {
    int m = i & 15, c = i >> 4;
    int l = l0 + m;
    sSeq[m * 193 + c] = x[(((size_t)b * C + c) * H + (l >> 6)) * W + (l & 63)];
  }
  __syncthreads();

  for (int t = 0; t < 2; ++t) {
    int n0 = (wv * 2 + t) * 16;
    v8f acc = gemm_tile(sG, 65, U_T, S, n0, R, lane);
#pragma unroll
    for (int v = 0; v < 8; ++v) {
      int row = v + 8 * g;
      size_t idx = ((size_t)(b * L + l0 + row)) * S + n0 + nn;
      float bx2 = bx_ws[idx] + bx_ws[(size_t)Bn * L * S + idx];
      sH[row * 129 + n0 + nn] = 0.5f * bx2 + 0.25f * acc[v];
    }
  }
  __syncthreads();

  for (int t = 0; t < 3; ++t) {
    int n0 = (wv * 3 + t) * 16;
    v8f yacc = gemm_tile(sH, 129, out_wT, C, n0, S, lane);
    v8f gacc = gemm_tile(sSeq, 193, gate_wT, C, n0, C, lane);
    float yb = out_b[n0 + nn], gb = gate_b[n0 + nn];
#pragma unroll
    for (int v = 0; v < 8; ++v) {
      int row = v + 8 * g;
      float yv = yacc[v] + yb;
      float gp = gacc[v] + gb;
      float si = gp / (1.f + __expf(-gp));
      sZ[row * 200 + n0 + nn] = si * yv + sSeq[row * 193 + n0 + nn];
    }
  }
  __syncthreads();

  {
    int row = tid >> 3, part = tid & 7;
    float s1 = 0.f, s2 = 0.f;
    for (int c = part * 24; c < part * 24 + 24; ++c) {
      float v = sZ[row * 200 + c];
      s1 += v; s2 += v * v;
    }
    red1[tid] = s1; red2[tid] = s2;
  }
  __syncthreads();
  if (tid < 16) {
    float s1 = 0.f, s2 = 0.f;
    for (int p = 0; p < 8; ++p) { s1 += red1[tid * 8 + p]; s2 += red2[tid * 8 + p]; }
    float mu = s1 / 192.f;
    float var = s2 / 192.f - mu * mu;
    stats[tid * 2] = mu;
    stats[tid * 2 + 1] = rsqrtf(var + 1e-5f);
  }
  __syncthreads();

  for (int i = tid; i < 16 * C; i += 128) {
    int m = i & 15, c = i >> 4;
    int l = l0 + m;
    float val = (sZ[m * 200 + c] - stats[m * 2]) * stats[m * 2 + 1] * norm_w[c] + norm_b[c];
    y_out[(((size_t)b * C + c) * H + (l >> 6)) * W + (l & 63)] = val;
  }
}

// ---------------------------------------------------------------------------
extern "C" void kernel_launch(void* const* d_in, const int* in_sizes, int n_in,
                              void* d_out, int out_size, void* d_ws, size_t ws_size,
                              hipStream_t stream) {
  const float* x      = (const float*)d_in[0];
  const float* in_w   = (const float*)d_in[1];
  const float* in_b   = (const float*)d_in[2];
  const float* U      = (const float*)d_in[3];
  const float* V      = (const float*)d_in[4];
  const float* B_w    = (const float*)d_in[5];
  const float* B_b    = (const float*)d_in[6];
  const float* out_w  = (const float*)d_in[7];
  const float* out_b  = (const float*)d_in[8];
  const float* norm_w = (const float*)d_in[9];
  const float* norm_b = (const float*)d_in[10];
  const float* gate_w = (const float*)d_in[11];
  const float* gate_b = (const float*)d_in[12];

  float* ws      = (float*)d_ws;
  float* in_wT   = ws;                    // [192][128]   24576
  float* B_wT    = in_wT + 24576;         // [128][128]   16384
  float* out_wT  = B_wT + 16384;          // [128][192]   24576
  float* gate_wT = out_wT + 24576;        // [192][192]   36864
  float* U_T     = gate_wT + 36864;       // [64][128]    8192
  float* Wt      = U_T + 8192;            // [64][64]     4096
  float* Pw      = Wt + 4096;             // 17 x [64][64] 69632
  float* Te      = Pw + 69632;            // [1024][64]   65536
  float* M2      = Te + 65536;            // [1088][1024] 1114112
  float* E       = M2 + 1114112;          // 4x[8][256][64] 524288
  float* Gq      = E + 524288;            // 4x[8][256][64] 524288
  float* bx      = Gq + 524288;           // 2x[8][4096][128] 8388608
  float* cbuf    = bx + 8388608;          // 2x[8][4096][64]  4194304
  float* g4      = cbuf + 4194304;        // 4x[8][4096][64]  8388608

  float* y_out     = (float*)d_out;                       // [8][192][64][64]
  float* state_out = y_out + (size_t)8 * 192 * 64 * 64;   // [8][4096][128]

  k_transpose<<<(24576 + 255) / 256, 256, 0, stream>>>(in_wT, in_w, 128, 192, 24576);
  k_transpose<<<(16384 + 255) / 256, 256, 0, stream>>>(B_wT, B_w, 128, 128, 16384);
  k_transpose<<<(24576 + 255) / 256, 256, 0, stream>>>(out_wT, out_w, 192, 128, 24576);
  k_transpose<<<(36864 + 255) / 256, 256, 0, stream>>>(gate_wT, gate_w, 192, 192, 36864);
  k_transpose<<<(8192 + 255) / 256, 256, 0, stream>>>(U_T, U, 128, 64, 8192);
  k_wt<<<16, 256, 0, stream>>>(U, V, Wt);
  k_powers<<<1, 256, 0, stream>>>(Wt, Pw);
  k_build_te<<<256, 256, 0, stream>>>(Pw, Te);
  k_build_m2<<<4352, 256, 0, stream>>>(Pw, M2);

  k_chain<<<dim3(256, 8, 2), 128, 0, stream>>>(x, in_wT, in_b, B_wT, B_b, V,
                                               state_out, bx, cbuf);
  k_carry_e<<<512, 128, 0, stream>>>(cbuf, Te, E);
  k_carry_scan<<<4, 128, 0, stream>>>(E, Pw + 16 * 4096, Gq);
  k_rebuild<<<512, 256, 0, stream>>>(cbuf, Gq, M2, g4);
  k_out<<<dim3(256, 8), 128, 0, stream>>>(x, bx, g4, U_T, out_wT, out_b,
                                          gate_wT, gate_b, norm_w, norm_b, y_out);
}